// OrderFreeLoss_10118942949455
// MI455X (gfx1250) — compile-verified
//
#include <hip/hip_runtime.h>
#include <hip/hip_bf16.h>
#include <stdint.h>

#define EOS_ID 2
#define S_LEN 32
#define B_SZ 128
#define V_SZ 32000
#define MAXL 256          // per-row label capacity (~20 expected; huge safety margin)
#define CHUNK 8000        // 32000 = 4 * 8000, 32 KB per LDS tile
#define NCHUNK 4

typedef unsigned int v4u __attribute__((ext_vector_type(4)));
typedef int          v8i __attribute__((ext_vector_type(8)));
typedef int          v4i __attribute__((ext_vector_type(4)));

// ---------------------------------------------------------------------------
// TDM: issue a 1-D tensor_load_to_lds of `nelem` 4-byte elements.
// D# per CDNA5 ISA §8: group0 = {count/flags, lds_addr, global_addr, type=2},
// group1 = {data_size=4B, tensor_dim0=nelem, tensor_dim1=1, tile_dim0=nelem,
//           tile_dim1=1, stride0=nelem}. Groups 2/3 zero (<=2D tensor).
// amdgpu-toolchain (clang-23) builtin arity: 6 args, trailing extra int32x8.
// ---------------------------------------------------------------------------
__device__ __forceinline__ void tdm_load_row(const void* gptr, void* lds_ptr, int nelem) {
    unsigned long long ga = (unsigned long long)(uintptr_t)gptr;
    // Generic pointers to LDS carry the LDS byte offset in their low 32 bits.
    unsigned int lds_off = (unsigned int)(uintptr_t)lds_ptr;
    v4u g0;
    g0.x = 1u;                                              // count=1 (valid), user mode
    g0.y = lds_off;                                         // lds_addr (bytes)
    g0.z = (unsigned int)(ga & 0xFFFFFFFFu);                // global_addr[31:0]
    g0.w = (unsigned int)((ga >> 32) & 0x01FFFFFFu)         // global_addr[56:32]
         | (2u << 30);                                      // type=2 ("image")
    v8i g1;
    g1[0] = (2 << 16);                                      // data_size=2 -> 4 bytes
    g1[1] = (nelem & 0xFFFF) << 16;                         // tensor_dim0[15:0] @ bits 63:48
    g1[2] = ((nelem >> 16) & 0xFFFF) | (1 << 16);           // tensor_dim0[31:16], tensor_dim1=1
    g1[3] = (nelem & 0xFFFF) << 16;                         // tile_dim0 @ bits 127:112
    g1[4] = 1;                                              // tile_dim1=1, tile_dim2=0
    g1[5] = nelem;                                          // tensor_dim0_stride[31:0]
    g1[6] = 0;                                              // stride0 hi / stride1 lo
    g1[7] = 0;                                              // stride1 hi
    v4i z4; z4[0] = 0; z4[1] = 0; z4[2] = 0; z4[3] = 0;
    v8i z8; z8[0] = 0; z8[1] = 0; z8[2] = 0; z8[3] = 0;
    z8[4] = 0; z8[5] = 0; z8[6] = 0; z8[7] = 0;
    __builtin_amdgcn_tensor_load_to_lds(g0, g1, z4, z4, z8, 0);   // TENSORcnt++
}

// ---------------------------------------------------------------------------
// K1: per-row label extraction. targets row (32000 f32) is streamed into LDS
// with double-buffered TDM tiles; 256 threads compact indices of entries >0.5.
// ---------------------------------------------------------------------------
__global__ __launch_bounds__(256) void k_extract_labels(
    const float* __restrict__ targets, int* __restrict__ labels,
    int* __restrict__ counts) {
    __shared__ float buf[2][CHUNK];
    __shared__ int lcount;
    const int b   = blockIdx.x;
    const int tid = threadIdx.x;
    const float* row = targets + (size_t)b * V_SZ;

    if (tid == 0) lcount = 0;
    if (tid < 32) {                                         // wave 0 drives the TDM
        tdm_load_row(row, &buf[0][0], CHUNK);               // tile 0 in flight
    }
    for (int c = 0; c < NCHUNK; ++c) {
        if (tid < 32) {
            if (c + 1 < NCHUNK)                             // overlap: issue next tile
                tdm_load_row(row + (size_t)(c + 1) * CHUNK, &buf[(c + 1) & 1][0], CHUNK);
            if (c + 1 < NCHUNK) __builtin_amdgcn_s_wait_tensorcnt(1); // tile c done
            else                __builtin_amdgcn_s_wait_tensorcnt(0);
        }
        __syncthreads();                                    // tile c visible to all waves
        const float* src = &buf[c & 1][0];
        const int base = c * CHUNK;
        for (int i = tid; i < CHUNK; i += 256) {
            if (src[i] > 0.5f) {
                int p = atomicAdd(&lcount, 1);
                if (p < MAXL) labels[b * MAXL + p] = base + i;
            }
        }
        __syncthreads();                                    // readers done before tile c+2 overwrites
    }
    if (tid == 0) counts[b] = (lcount < MAXL) ? lcount : MAXL;
}

// monotone float -> u32 (order preserving), and its inverse
__device__ __forceinline__ unsigned f2key(float f) {
    unsigned u = __float_as_uint(f);
    return (u & 0x80000000u) ? ~u : (u | 0x80000000u);
}
__device__ __forceinline__ float key2f(unsigned u) {
    return __uint_as_float((u & 0x80000000u) ? (u & 0x7FFFFFFFu) : ~u);
}

// ---------------------------------------------------------------------------
// K2: one wave32 per batch row. 32-step greedy decode over <=MAXL active
// labels (8 slots/lane). Wave argmax via 64-bit key: (f2key(val)<<32)|~idx
// so max key == max value with ties broken toward the smallest index,
// exactly matching jnp.argmax over exp(out)*tgts.
// ---------------------------------------------------------------------------
__global__ __launch_bounds__(32) void k_decode(
    const float* __restrict__ outputs, const int* __restrict__ labels,
    const int* __restrict__ counts, float* __restrict__ partial) {
    const int b = blockIdx.x;
    const int lane = threadIdx.x;
    int cnt = counts[b]; if (cnt > MAXL) cnt = MAXL;

    int  idxs[MAXL / 32];
    bool act [MAXL / 32];
    #pragma unroll
    for (int k = 0; k < MAXL / 32; ++k) {
        int s = lane + 32 * k;
        act[k]  = (s < cnt);
        idxs[k] = act[k] ? labels[b * MAXL + s] : 0;
    }

    bool eos = false;
    int  rem = cnt;            // total active entries (labels + EOS flag)
    float acc = 0.0f;          // sum of chosen log-probs (uniform across lanes)
    const float* outb = outputs + (size_t)b * V_SZ;

    for (int t = 0; t < S_LEN; ++t) {
        const float* outt = outb + (size_t)t * B_SZ * V_SZ;
        unsigned long long key = 0ull;                      // "no candidate"
        #pragma unroll
        for (int k = 0; k < MAXL / 32; ++k) {
            if (act[k]) {
                unsigned u = f2key(outt[idxs[k]]);
                unsigned long long kk =
                    ((unsigned long long)u << 32) | (unsigned)~(unsigned)idxs[k];
                if (kk > key) key = kk;
                if (t + 1 < S_LEN)                          // gfx1250 global_prefetch_b8
                    __builtin_prefetch(outt + (size_t)B_SZ * V_SZ + idxs[k], 0, 0);
            }
        }
        if (eos && lane == 0) {
            unsigned u = f2key(outt[EOS_ID]);
            unsigned long long kk =
                ((unsigned long long)u << 32) | (unsigned)~(unsigned)EOS_ID;
            if (kk > key) key = kk;
        }
        // wave32 max-reduce of the packed key
        #pragma unroll
        for (int off = 16; off > 0; off >>= 1) {
            unsigned long long o = __shfl_xor(key, off, 32);
            if (o > key) key = o;
        }
        unsigned widx = ~(unsigned)key;
        float    wval = key2f((unsigned)(key >> 32));
        acc += wval;
        if ((int)widx == EOS_ID) {
            eos = false;                                    // EOS consumed this step
        } else {
            #pragma unroll
            for (int k = 0; k < MAXL / 32; ++k)
                if (act[k] && (unsigned)idxs[k] == widx) act[k] = false;
        }
        rem -= 1;
        if (rem == 0) { eos = true; rem = 1; }              // tgts.at[:,EOS].add(is_end)
    }
    if (lane == 0) partial[b] = -acc;
}

// ---------------------------------------------------------------------------
// K3: deterministic tree reduction of 128 per-row partial sums; mean over S*B.
// ---------------------------------------------------------------------------
__global__ __launch_bounds__(128) void k_finalize(
    const float* __restrict__ partial, float* __restrict__ out) {
    __shared__ float s[128];
    int t = threadIdx.x;
    s[t] = partial[t];
    __syncthreads();
    for (int off = 64; off > 0; off >>= 1) {
        if (t < off) s[t] += s[t + off];
        __syncthreads();
    }
    if (t == 0) out[0] = s[0] / (float)(S_LEN * B_SZ);
}

extern "C" void kernel_launch(void* const* d_in, const int* in_sizes, int n_in,
                              void* d_out, int out_size, void* d_ws, size_t ws_size,
                              hipStream_t stream) {
    (void)in_sizes; (void)n_in; (void)out_size; (void)ws_size;
    const float* outputs = (const float*)d_in[0];           // [S,B,V] f32 log-probs
    // d_in[1] = output_symbols: stacked but unused in the executed branch
    const float* targets = (const float*)d_in[2];           // [B,V] f32 multi-hot

    char* ws = (char*)d_ws;
    int*   labels  = (int*)  (ws);                          // B*MAXL ints  (128 KB)
    int*   counts  = (int*)  (ws + (size_t)B_SZ * MAXL * 4);//  B ints
    float* partial = (float*)(ws + (size_t)B_SZ * MAXL * 4 + B_SZ * 4);

    k_extract_labels<<<B_SZ, 256, 0, stream>>>(targets, labels, counts);
    k_decode        <<<B_SZ,  32, 0, stream>>>(outputs, labels, counts, partial);
    k_finalize      <<<1,    128, 0, stream>>>(partial, (float*)d_out);
}